// Harris_35751307772666
// MI455X (gfx1250) — compile-verified
//
#include <hip/hip_runtime.h>
#include <math.h>

// Harris corner pipeline, fully fused, LDS-staged, async global->LDS copies.
// Image: 1 x 3 x 4096 x 4096 f32. Outputs: edge(16M) ++ eig(16M) f32.
// Memory-bound: ~192 MB in + 128 MB out => ~14 us roofline at 23.3 TB/s.

#define IMG_W 4096
#define IMG_H 4096
#define TILE  32
#define CH_COLS 40           // vec4-aligned halo row: gx in [tx0-4, tx0+36)
#define CH_ROWS 36           // gy in [ty0-2, ty0+34)
#define CH_PER  (CH_ROWS * CH_COLS)      // 1440 floats per channel
#define G_STRIDE 37          // 36 gray cols + 1 pad
#define E_STRIDE 35          // 34 grad cols + 1 pad

__device__ __forceinline__ void async_ld_b128(unsigned lds_off, const float* src) {
    // GLOBAL_LOAD_ASYNC_TO_LDS_B128 (GV mode): vdst = LDS byte offset,
    // vaddr = 64-bit global address. Tracked by ASYNCcnt.
    asm volatile("global_load_async_to_lds_b128 %0, %1, off"
                 :: "v"(lds_off), "v"(src) : "memory");
}

__global__ __launch_bounds__(256) void harris_fused(const float* __restrict__ x,
                                                    float* __restrict__ out)
{
    __shared__ __align__(16) float chS[3 * CH_PER];    // raw channel halo tiles
    __shared__ float grayS[CH_ROWS * G_STRIDE];        // 36x36 gray halo
    __shared__ float IxS[34 * E_STRIDE];               // 34x34 Scharr-x
    __shared__ float IyS[34 * E_STRIDE];               // 34x34 Scharr-y

    const int tid = threadIdx.x;
    const int tx0 = blockIdx.x * TILE;
    const int ty0 = blockIdx.y * TILE;
    const bool interior = (tx0 >= 4) & (tx0 + 36 <= IMG_W) &
                          (ty0 >= 2) & (ty0 + 34 <= IMG_H);

    // ---------------- Stage 1: async DMA of 3-channel halo tile into LDS -----
    // 10 vec4 chunks per row (40 floats), 36 rows, 3 channels = 1080 chunks.
    if (interior) {
        for (int idx = tid; idx < 3 * CH_ROWS * 10; idx += 256) {
            int c   = idx / (CH_ROWS * 10);
            int r   = idx - c * (CH_ROWS * 10);
            int ly  = r / 10;
            int vx  = r - ly * 10;
            const float* src = x + ((size_t)c * IMG_H + (ty0 - 2 + ly)) * IMG_W
                                 + (tx0 - 4 + (vx << 2));
            unsigned lds = (unsigned)(unsigned long long)
                           (&chS[c * CH_PER + ly * CH_COLS + (vx << 2)]);
            async_ld_b128(lds, src);
        }
    } else {
        for (int idx = tid; idx < 3 * CH_ROWS * 10; idx += 256) {
            int c   = idx / (CH_ROWS * 10);
            int r   = idx - c * (CH_ROWS * 10);
            int ly  = r / 10;
            int vx  = r - ly * 10;
            int gy  = ty0 - 2 + ly;
            int gx4 = tx0 - 4 + (vx << 2);              // 16B aligned (tx0 % 32 == 0)
            int off = c * CH_PER + ly * CH_COLS + (vx << 2);
            if ((unsigned)gy < (unsigned)IMG_H && (unsigned)gx4 < (unsigned)IMG_W) {
                const float* src = x + ((size_t)c * IMG_H + gy) * IMG_W + gx4;
                async_ld_b128((unsigned)(unsigned long long)(&chS[off]), src);
            } else {
                float4 z = {0.f, 0.f, 0.f, 0.f};
                *reinterpret_cast<float4*>(&chS[off]) = z;  // zero padding
            }
        }
    }
    asm volatile("s_wait_asynccnt 0x0" ::: "memory");
    __syncthreads();

    // ---------------- Stage 2: grayscale (36x36 halo) ------------------------
    for (int idx = tid; idx < CH_ROWS * 36; idx += 256) {
        int ly = idx / 36;
        int lx = idx - ly * 36;
        int co = ly * CH_COLS + lx + 2;                 // halo col 0 == chS col 2
        float g = 0.299f * chS[co]
                + 0.587f * chS[co + CH_PER]
                + 0.114f * chS[co + 2 * CH_PER];
        grayS[ly * G_STRIDE + lx] = g;
    }
    __syncthreads();

    // ---------------- Stage 3: Scharr gradients (34x34 inner grid) -----------
    for (int idx = tid; idx < 34 * 34; idx += 256) {
        int py = idx / 34;
        int px = idx - py * 34;
        float vx = 0.f, vy = 0.f;
        bool live = interior;
        if (!interior) {
            int iy = ty0 - 1 + py;                      // image coords of this grad
            int ix = tx0 - 1 + px;
            live = ((unsigned)iy < (unsigned)IMG_H) && ((unsigned)ix < (unsigned)IMG_W);
        }
        if (live) {
            const float* g0 = &grayS[py * G_STRIDE + px];   // top-left of 3x3
            float tl = g0[0],            tc = g0[1],            tr = g0[2];
            float ml = g0[G_STRIDE],                            mr = g0[G_STRIDE + 2];
            float bl = g0[2 * G_STRIDE], bc = g0[2 * G_STRIDE + 1], br = g0[2 * G_STRIDE + 2];
            vx = 3.f * (tr - tl) + 10.f * (mr - ml) + 3.f * (br - bl);
            vy = 3.f * (bl - tl) + 10.f * (bc - tc) + 3.f * (br - tr);
        }
        IxS[py * E_STRIDE + px] = vx;   // zero outside image -> box conv zero pad
        IyS[py * E_STRIDE + px] = vy;
    }
    __syncthreads();

    // ---------------- Stage 4: box sums + Harris response, 4 px/thread -------
    const int ox  = tid & 31;
    const int oyb = tid >> 5;
#pragma unroll
    for (int k = 0; k < 4; ++k) {
        int oy = oyb + (k << 3);
        int cy = oy + 1, cx = ox + 1;                   // center in grad grid
        float sA = 0.f, sB = 0.f, sC = 0.f;
#pragma unroll
        for (int dy = -1; dy <= 1; ++dy) {
#pragma unroll
            for (int dx = -1; dx <= 1; ++dx) {
                float a = IxS[(cy + dy) * E_STRIDE + cx + dx];
                float b = IyS[(cy + dy) * E_STRIDE + cx + dx];
                sA = fmaf(a, a, sA);
                sB = fmaf(b, b, sB);
                sC = fmaf(a, b, sC);
            }
        }
        float cIx = IxS[cy * E_STRIDE + cx];
        float cIy = IyS[cy * E_STRIDE + cx];
        float edge  = 0.5f * (fabsf(cIx) + fabsf(cIy));
        float diff  = sA - sB;
        float trace = sA + sB;
        float eig   = trace - sqrtf(fmaf(diff, diff, 4.f * sC * sC));
        size_t o = (size_t)(ty0 + oy) * IMG_W + (tx0 + ox);
        // Outputs are write-once streams (128 MB): non-temporal so they do not
        // evict input halo lines from L2 (input rows are reused by the tile row below).
        __builtin_nontemporal_store(edge, &out[o]);
        __builtin_nontemporal_store(eig,  &out[(size_t)IMG_W * IMG_H + o]);
    }
}

extern "C" void kernel_launch(void* const* d_in, const int* in_sizes, int n_in,
                              void* d_out, int out_size, void* d_ws, size_t ws_size,
                              hipStream_t stream) {
    const float* x = (const float*)d_in[0];
    float* out = (float*)d_out;
    dim3 grid(IMG_W / TILE, IMG_H / TILE);
    harris_fused<<<grid, dim3(256), 0, stream>>>(x, out);
}